// MultiHeadAttention_66812511256972
// MI455X (gfx1250) — compile-verified
//
#include <hip/hip_runtime.h>
#include <hip/hip_bf16.h>
#include <math.h>
#include <stdint.h>

// ---------------- problem constants ----------------
#define B_    8
#define LQ_   1024
#define LK_   1024
#define H_    8
#define HD_   128
#define HHD_  1024          // H_*HD_
#define HEADDIM_ 512
#define DM_   1024          // D_MODEL
#define NEG_  100000000.0f
#define EPS_  1e-5f
#define INV_TEMP_ 0.0883883476483184f   // 1/sqrt(128)

#define KC_   32            // K-chunk per LDS stage
#define LDSP_ 36            // padded row stride (floats): 144B rows -> 16B aligned,
                            // and 36*n mod 64 banks is a permutation for n=0..15

typedef float v2f __attribute__((ext_vector_type(2)));
typedef float v8f __attribute__((ext_vector_type(8)));

// ---------------- CDNA5 async copy global->LDS (ASYNCcnt) ----------------
__device__ __forceinline__ void async_ld16(uint32_t lds_byte_off, const void* gptr) {
    asm volatile("global_load_async_to_lds_b128 %0, %1, off"
                 :: "v"(lds_byte_off), "v"(gptr) : "memory");
}
template <int N>
__device__ __forceinline__ void wait_async() {
    asm volatile("s_wait_asynccnt %0" :: "i"(N) : "memory");
}
__device__ __forceinline__ uint32_t lds_off32(const void* p) {
    return (uint32_t)(uintptr_t)p;   // low 32 bits of shared-aperture flat addr = LDS offset
}

// wave32 reductions
__device__ __forceinline__ float wave_max(float v) {
#pragma unroll
    for (int off = 16; off > 0; off >>= 1) v = fmaxf(v, __shfl_xor(v, off, 32));
    return v;
}
__device__ __forceinline__ float wave_sum(float v) {
#pragma unroll
    for (int off = 16; off > 0; off >>= 1) v += __shfl_xor(v, off, 32);
    return v;
}

// ---------------------------------------------------------------------------
// Double-buffered LDS GEMM core: acc[NT] += A[M x K] @ W[N x K]^T  (block tile)
//   BM = WM*16, BN = WN*NT*16, K % 32 == 0
//   Al: float[2][BM*LDSP_]   Wl: float[2][BN*LDSP_]   (shared)
// f32 WMMA 16x16x4 fragment layout:
//   lanes 0-15: K = k+0,k+1 ; lanes 16-31: K = k+2,k+3 (both A rows and W cols)
// ---------------------------------------------------------------------------
template <int WM, int WN, int NT>
__device__ __forceinline__ void gemm_core(
    const float* __restrict__ A, int lda,
    const float* __restrict__ W, int ldw,
    int K, int mblk, int nblk,
    float* __restrict__ Al, float* __restrict__ Wl,
    v8f acc[NT])
{
    constexpr int BM  = WM * 16;
    constexpr int BN  = WN * NT * 16;
    constexpr int AQ  = BM * 8;               // 16B quanta in A tile
    constexpr int WQ  = BN * 8;               // 16B quanta in W tile
    constexpr int PER = (AQ + WQ) / 256;      // async insts per thread per chunk

    const int tid  = threadIdx.x;
    const int lane = tid & 31;
    const int wave = tid >> 5;
    const int wm = wave % WM, wn = wave / WM;
    const int lm = lane & 15;
    const int khalf = (lane >> 4) << 1;       // 0 or 2

    auto issue = [&](int buf, int kc) {
#pragma unroll
        for (int i = tid; i < AQ; i += 256) {
            const int row = i >> 3, qq = i & 7;
            async_ld16(lds_off32(&Al[buf * (BM * LDSP_) + row * LDSP_ + qq * 4]),
                       A + (size_t)(mblk + row) * lda + kc + qq * 4);
        }
#pragma unroll
        for (int i = tid; i < WQ; i += 256) {
            const int row = i >> 3, qq = i & 7;
            async_ld16(lds_off32(&Wl[buf * (BN * LDSP_) + row * LDSP_ + qq * 4]),
                       W + (size_t)(nblk + row) * ldw + kc + qq * 4);
        }
    };

    const int nchunks = K / KC_;
    issue(0, 0);
    for (int c = 0; c < nchunks; ++c) {
        if (c + 1 < nchunks) {
            issue((c + 1) & 1, (c + 1) * KC_);
            wait_async<PER>();               // chunk c complete (in-order)
        } else {
            wait_async<0>();
        }
        __syncthreads();                      // chunk c visible to all waves

        const float* Ab = &Al[(c & 1) * (BM * LDSP_)];
        const float* Wb = &Wl[(c & 1) * (BN * LDSP_)];
#pragma unroll
        for (int kk = 0; kk < KC_; kk += 4) {
            v2f a = *(const v2f*)&Ab[(wm * 16 + lm) * LDSP_ + kk + khalf];
#pragma unroll
            for (int t = 0; t < NT; ++t) {
                v2f w = *(const v2f*)&Wb[(wn * (NT * 16) + t * 16 + lm) * LDSP_ + kk + khalf];
                acc[t] = __builtin_amdgcn_wmma_f32_16x16x4_f32(
                    false, a, false, w, (short)0, acc[t], false, false);
            }
        }
        __syncthreads();                      // done reading before buffer reuse
    }
}

// ---------------------------------------------------------------------------
// Generic GEMM:  C = A @ W^T + bias (+ residual), WM=2 WN=4 NT=4 (32 x 256)
// ---------------------------------------------------------------------------
template <bool TRANS_OUT>
__global__ __launch_bounds__(256)
void gemm_f32_wmma(const float* __restrict__ A, int lda,
                   const float* __restrict__ W, int ldw,
                   const float* __restrict__ bias,
                   const float* __restrict__ res,
                   float* __restrict__ C, int ldc,
                   int K)
{
    __shared__ float Al[2 * 32 * LDSP_];
    __shared__ float Wl[2 * 256 * LDSP_];

    const int lane = threadIdx.x & 31;
    const int wave = threadIdx.x >> 5;
    const int wm = wave % 2, wn = wave / 2;
    const int mblk = blockIdx.x * 32;
    const int nblk = blockIdx.y * 256;
    const int lm = lane & 15;

    v8f acc[4] = {};
    gemm_core<2, 4, 4>(A, lda, W, ldw, K, mblk, nblk, Al, Wl, acc);

    const int mb = mblk + wm * 16 + ((lane >> 4) << 3);
#pragma unroll
    for (int t = 0; t < 4; ++t) {
        const int n = nblk + wn * 64 + t * 16 + lm;
        const float bv = bias ? bias[n] : 0.0f;
#pragma unroll
        for (int r = 0; r < 8; ++r) {
            float val = acc[t][r] + bv;
            size_t idx = TRANS_OUT ? ((size_t)n * ldc + (mb + r))
                                   : ((size_t)(mb + r) * ldc + n);
            if (res) val += res[idx];
            C[idx] = val;
        }
    }
}

// ---------------------------------------------------------------------------
// Attention scores (z = h*B+b):  raw = qh_bh @ kh_bh^T + old ; sbuf = masked/scaled
// WM=2 WN=4 NT=4 (32 x 256), K = 128
// ---------------------------------------------------------------------------
__global__ __launch_bounds__(256)
void attn_scores_kernel(const float* __restrict__ qh, const float* __restrict__ kh,
                        const float* __restrict__ old_score,
                        const float* __restrict__ fmask,
                        float* __restrict__ raw_out, float* __restrict__ sbuf)
{
    __shared__ float Al[2 * 32 * LDSP_];
    __shared__ float Wl[2 * 256 * LDSP_];

    const int z = blockIdx.z;
    const int h = z / B_, b = z % B_;
    const int lane = threadIdx.x & 31;
    const int wave = threadIdx.x >> 5;
    const int wm = wave % 2, wn = wave / 2;
    const int mblk = blockIdx.x * 32;
    const int nblk = blockIdx.y * 256;
    const int lm = lane & 15;

    const float* A  = qh + (size_t)b * LQ_ * HHD_ + (size_t)h * HD_;
    const float* Wp = kh + (size_t)b * LK_ * HHD_ + (size_t)h * HD_;

    v8f acc[4] = {};
    gemm_core<2, 4, 4>(A, HHD_, Wp, HHD_, HD_, mblk, nblk, Al, Wl, acc);

    const size_t zoff = (size_t)z * LQ_ * LK_;
    const size_t boff = (size_t)b * LQ_ * LK_;
    const int mb = mblk + wm * 16 + ((lane >> 4) << 3);
#pragma unroll
    for (int t = 0; t < 4; ++t) {
        const int n = nblk + wn * 64 + t * 16 + lm;
#pragma unroll
        for (int r = 0; r < 8; ++r) {
            const int m = mb + r;
            const size_t idx = zoff + (size_t)m * LK_ + n;
            float raw = acc[t][r] + old_score[idx];
            raw_out[idx] = raw;
            float fm = fmask[boff + (size_t)m * LK_ + n];
            sbuf[idx] = (raw - NEG_ * (1.0f - fm)) * INV_TEMP_;
        }
    }
}

// ---------------------------------------------------------------------------
// attn @ V (z = h*B+b): out[b,:,h*128:+128] = attn[z] @ vh_t[h*128:+128, b*L:+L]^T
// WM=4 WN=2 NT=4 (64 x 128), K = 1024, vh_t is [HHD_][B*L] (ldw = 8192)
// ---------------------------------------------------------------------------
__global__ __launch_bounds__(256)
void attn_v_kernel(const float* __restrict__ attn, const float* __restrict__ vh_t,
                   float* __restrict__ outh)
{
    __shared__ float Al[2 * 64 * LDSP_];
    __shared__ float Wl[2 * 128 * LDSP_];

    const int z = blockIdx.z;
    const int h = z / B_, b = z % B_;
    const int lane = threadIdx.x & 31;
    const int wave = threadIdx.x >> 5;
    const int wm = wave % 4, wn = wave / 4;
    const int mblk = blockIdx.x * 64;
    const int lm = lane & 15;
    const int ldw = B_ * LQ_;                 // 8192

    const float* A  = attn + (size_t)z * LQ_ * LK_;
    const float* Wp = vh_t + (size_t)h * HD_ * ldw + (size_t)b * LQ_;

    v8f acc[4] = {};
    gemm_core<4, 2, 4>(A, LK_, Wp, ldw, LK_, mblk, 0, Al, Wl, acc);

    const int mb = mblk + wm * 16 + ((lane >> 4) << 3);
#pragma unroll
    for (int t = 0; t < 4; ++t) {
        const int n = wn * 64 + t * 16 + lm;
#pragma unroll
        for (int r = 0; r < 8; ++r) {
            const int m = mb + r;
            outh[(size_t)b * LQ_ * HHD_ + (size_t)m * HHD_ + h * HD_ + n] = acc[t][r];
        }
    }
}

// ---------------------------------------------------------------------------
// Row softmax in-place over rows of length 1024 (one block / row, 256 threads)
// ---------------------------------------------------------------------------
__global__ __launch_bounds__(256)
void softmax_rows_kernel(float* __restrict__ s)
{
    float* p = s + (size_t)blockIdx.x * LK_;
    const int tid = threadIdx.x;
    const int w = tid >> 5, lane = tid & 31;
    __shared__ float sm[8], ss[8];

    float x[4];
#pragma unroll
    for (int j = 0; j < 4; ++j) x[j] = p[tid + 256 * j];

    float m = fmaxf(fmaxf(x[0], x[1]), fmaxf(x[2], x[3]));
    m = wave_max(m);
    if (lane == 0) sm[w] = m;
    __syncthreads();
    float bm = sm[0];
#pragma unroll
    for (int i = 1; i < 8; ++i) bm = fmaxf(bm, sm[i]);

    float e[4], lsum = 0.0f;
#pragma unroll
    for (int j = 0; j < 4; ++j) { e[j] = __expf(x[j] - bm); lsum += e[j]; }
    lsum = wave_sum(lsum);
    if (lane == 0) ss[w] = lsum;
    __syncthreads();
    float bs = 0.0f;
#pragma unroll
    for (int i = 0; i < 8; ++i) bs += ss[i];
    float inv = 1.0f / bs;
#pragma unroll
    for (int j = 0; j < 4; ++j) p[tid + 256 * j] = e[j] * inv;
}

// ---------------------------------------------------------------------------
// Row LayerNorm: rows of 1024 (one block / row, 256 threads)
// ---------------------------------------------------------------------------
__global__ __launch_bounds__(256)
void layernorm_rows_kernel(const float* __restrict__ fc,
                           const float* __restrict__ g, const float* __restrict__ bb,
                           float* __restrict__ out)
{
    const float* p = fc + (size_t)blockIdx.x * DM_;
    float* o = out + (size_t)blockIdx.x * DM_;
    const int tid = threadIdx.x;
    const int w = tid >> 5, lane = tid & 31;
    __shared__ float s1[8], s2[8];

    float x[4];
    float sum = 0.0f, sq = 0.0f;
#pragma unroll
    for (int j = 0; j < 4; ++j) {
        x[j] = p[tid + 256 * j];
        sum += x[j];
        sq += x[j] * x[j];
    }
    sum = wave_sum(sum);
    sq  = wave_sum(sq);
    if (lane == 0) { s1[w] = sum; s2[w] = sq; }
    __syncthreads();
    float bsum = 0.0f, bsq = 0.0f;
#pragma unroll
    for (int i = 0; i < 8; ++i) { bsum += s1[i]; bsq += s2[i]; }
    const float mean = bsum * (1.0f / DM_);
    const float var  = bsq * (1.0f / DM_) - mean * mean;
    const float inv  = rsqrtf(var + EPS_);
#pragma unroll
    for (int j = 0; j < 4; ++j) {
        const int n = tid + 256 * j;
        o[n] = (x[j] - mean) * inv * g[n] + bb[n];
    }
}

// ---------------------------------------------------------------------------
extern "C" void kernel_launch(void* const* d_in, const int* in_sizes, int n_in,
                              void* d_out, int out_size, void* d_ws, size_t ws_size,
                              hipStream_t stream)
{
    const float* q    = (const float*)d_in[0];
    const float* k    = (const float*)d_in[1];
    const float* v    = (const float*)d_in[2];
    const float* olds = (const float*)d_in[3];
    const float* fm   = (const float*)d_in[4];
    const float* wq   = (const float*)d_in[5];
    const float* bq   = (const float*)d_in[6];
    const float* wk   = (const float*)d_in[7];
    const float* bk   = (const float*)d_in[8];
    const float* wv   = (const float*)d_in[9];
    const float* bv   = (const float*)d_in[10];
    const float* wfc  = (const float*)d_in[11];
    const float* bfc  = (const float*)d_in[12];
    const float* lng  = (const float*)d_in[13];
    const float* lnb  = (const float*)d_in[14];

    float* out = (float*)d_out;                          // [B,LQ,DM]  8M floats
    float* raw = out + (size_t)B_ * LQ_ * DM_;           // [H*B,LQ,LK] 64M floats

    const size_t M8 = (size_t)8 * 1024 * 1024;
    float* ws   = (float*)d_ws;
    float* qh   = ws;            // [B*LQ, 1024]
    float* kh   = qh + M8;       // [B*LK, 1024]
    float* vh_t = kh + M8;       // [1024, B*LK]   (transposed store)
    float* outh = vh_t + M8;     // [B*LQ, 1024]
    float* fc   = outh + M8;     // [B*LQ, 1024]
    float* attn = fc + M8;       // [H*B, LQ, LK]  64M floats

    const dim3 blk(256);
    const int MROWS = B_ * LQ_;                          // 8192

    // Q / K projections: [8192,512] @ [1024,512]^T -> [8192,1024]
    gemm_f32_wmma<false><<<dim3(MROWS / 32, 4), blk, 0, stream>>>(
        q, HEADDIM_, wq, HEADDIM_, bq, nullptr, qh, HHD_, HEADDIM_);
    gemm_f32_wmma<false><<<dim3(MROWS / 32, 4), blk, 0, stream>>>(
        k, HEADDIM_, wk, HEADDIM_, bk, nullptr, kh, HHD_, HEADDIM_);
    // V projection, stored transposed: vh_t[d][row] (ldc = 8192)
    gemm_f32_wmma<true><<<dim3(MROWS / 32, 4), blk, 0, stream>>>(
        v, DM_, wv, DM_, bv, nullptr, vh_t, MROWS, DM_);

    // scores + raw output + mask/scale
    attn_scores_kernel<<<dim3(LQ_ / 32, LK_ / 256, H_ * B_), blk, 0, stream>>>(
        qh, kh, olds, fm, raw, attn);

    // softmax over each of the H*B*LQ rows
    softmax_rows_kernel<<<dim3(H_ * B_ * LQ_), blk, 0, stream>>>(attn);

    // attn @ V
    attn_v_kernel<<<dim3(LQ_ / 64, 1, H_ * B_), blk, 0, stream>>>(attn, vh_t, outh);

    // FC + bias + residual(v)
    gemm_f32_wmma<false><<<dim3(MROWS / 32, 4), blk, 0, stream>>>(
        outh, HHD_, wfc, HHD_, bfc, v, fc, DM_, HHD_);

    // LayerNorm -> final out
    layernorm_rows_kernel<<<dim3(MROWS), blk, 0, stream>>>(fc, lng, lnb, out);
}